// Interpolator_8693013807470
// MI455X (gfx1250) — compile-verified
//
#include <hip/hip_runtime.h>
#include <math.h>

// ---------------------------------------------------------------------------
// Bilinear keypoint interpolation (B=4, C=128, H=W=256, N=4096), f32 exact.
//
// out[b,c,n] = mask(n) * sum_{k<4} feature[b,c, iy(n,k)*W+ix(n,k)] * w(n,k)
//
// Memory-bound (~42 MB traffic, feature fits in 192MB L2). The k=4 corner
// contraction is mapped onto V_WMMA_F32_16X16X4_F32 (K==4): per 16-channel x
// 16-keypoint tile we chain 16 WMMAs, iteration j supplying keypoint j's
// gathered corners as A (16 channels x 4 corners) and a one-hot column of its
// 4 mask-scaled bilinear weights as B, accumulating through SRC2. Full f32.
// ---------------------------------------------------------------------------

typedef __attribute__((ext_vector_type(2))) float v2f;
typedef __attribute__((ext_vector_type(8))) float v8f;

__global__ __launch_bounds__(256) void interp_wmma_kernel(
    const float* __restrict__ feature,    // B x C x H x W
    const float* __restrict__ keypoints,  // B x N x 2  (x, y) in image pixels
    float* __restrict__ out)              // B x C x N
{
    constexpr int C = 128, H = 256, W = 256, Npts = 4096;
    constexpr float INV_RATIO = 1.0f / 16.0f;   // 1 / IM_FE_RATIO (exact pow2)

    const int b     = blockIdx.y;
    const int nbase = blockIdx.x << 4;          // 16 keypoints per block
    const int wave  = threadIdx.x >> 5;         // 8 waves -> 8 channel tiles
    const int lane  = threadIdx.x & 31;
    const int half  = lane >> 4;                // 0: corners {k0,k1}, 1: {k2,k3}
    const int m     = lane & 15;                // A-row (channel) & B-col (kp)

    // This lane's feature row: channel = wave*16 + m  (A-matrix row M=m).
    const float* __restrict__ frow =
        feature + ((size_t)(b * C + (wave << 4) + m) << 16);   // * H*W
    // Block-uniform keypoint slice -> scalar loads.
    const float* __restrict__ kp = keypoints + (size_t)(b * Npts + nbase) * 2;

    v8f acc = {};   // 16x16 f32 C/D accumulator (8 VGPRs)

#pragma unroll
    for (int j = 0; j < 16; ++j) {
        // --- bilinear setup for keypoint (nbase + j), matches reference ---
        const float kx = kp[2 * j]     * INV_RATIO;
        const float ky = kp[2 * j + 1] * INV_RATIO;
        float x0f = floorf(kx); x0f = x0f < 0.0f ? 0.0f : x0f;
        float y0f = floorf(ky); y0f = y0f < 0.0f ? 0.0f : y0f;
        float x1f = ceilf(kx);  x1f = x1f > (float)(W - 1) ? (float)(W - 1) : x1f;
        float y1f = ceilf(ky);  y1f = y1f > (float)(H - 1) ? (float)(H - 1) : y1f;
        const float ux = kx - x0f, uy = ky - y0f;     // upper-corner weights
        const float lx = 1.0f - ux, ly = 1.0f - uy;   // lower-corner weights
        // mask = mean over (x,y) of (kp > 1e-10); fold into weights (linear).
        const float mask = 0.5f * ((kx > 1e-10f ? 1.0f : 0.0f) +
                                   (ky > 1e-10f ? 1.0f : 0.0f));
        const int x0 = (int)x0f, x1 = (int)x1f;
        const int y0 = (int)y0f, y1 = (int)y1f;

        // Lane-half selects its corner pair (branchless, EXEC stays all-1):
        //   half 0 -> k0=(x0,y0), k1=(x1,y0);  half 1 -> k2=(x0,y1), k3=(x1,y1)
        const int   yrow = half ? y1 : y0;
        const float wy   = half ? uy : ly;
        const int   iA   = yrow * W + x0;
        const int   iB   = yrow * W + x1;
        const float wA   = lx * wy * mask;
        const float wB   = ux * wy * mask;

        // A: 16x4 f32. lanes 0-15: (M=lane, K0|K1); lanes 16-31: (M, K2|K3).
        // Two corners on one y-row share a cacheline (x1 == x0+1 typically).
        v2f a;
        a.x = frow[iA];
        a.y = frow[iB];

        // B: 4x16 f32, one-hot column j (assumed lane-half-symmetric layout:
        // V0 = rows K0|K2, V1 = rows K1|K3, N = lane%16).
        const bool sel = (m == j);
        v2f bv;
        bv.x = sel ? wA : 0.0f;
        bv.y = sel ? wB : 0.0f;

        // D = A x B + C  -> fills output column j of the 16x16 tile.
        acc = __builtin_amdgcn_wmma_f32_16x16x4_f32(
            false, a, false, bv, (short)0, acc, false, false);
    }

    // C/D layout: VGPR r -> M = r (lanes 0-15) / 8+r (lanes 16-31); N = m.
    // Output is write-once -> non-temporal stores, keep L2 for the feature.
    float* __restrict__ orow =
        out + (size_t)(b * C + (wave << 4) + (half << 3)) * Npts + nbase + m;
#pragma unroll
    for (int r = 0; r < 8; ++r)
        __builtin_nontemporal_store(acc[r], &orow[(size_t)r * Npts]);
}

extern "C" void kernel_launch(void* const* d_in, const int* in_sizes, int n_in,
                              void* d_out, int out_size, void* d_ws, size_t ws_size,
                              hipStream_t stream) {
    const float* feature   = (const float*)d_in[0];   // 4*128*256*256 f32
    const float* keypoints = (const float*)d_in[1];   // 4*4096*2 f32
    float* out             = (float*)d_out;           // 4*128*4096 f32

    constexpr int B = 4, Npts = 4096;
    dim3 grid(Npts / 16, B);   // (keypoint tile, batch)
    dim3 block(256);           // 8 waves x 16 channels = C=128
    hipLaunchKernelGGL(interp_wmma_kernel, grid, block, 0, stream,
                       feature, keypoints, out);
}